// FinDiffScheme_30640296690423
// MI455X (gfx1250) — compile-verified
//
#include <hip/hip_runtime.h>
#include <hip/hip_bf16.h>
#include <stdint.h>

// Problem sizes fixed by the reference: (num_t, num_x, B) = (129, 2050, 128)
#define NXI    2048   // interior mesh points
#define NSTEP  128    // time steps
#define NBATCH 128    // batch rows (one workgroup each)
#define NTH    256    // threads per workgroup = 8 wave32
#define EPT    8      // elements per thread (NXI / NTH)
#define DCOEF  0.13f  // VOL^2 + VOL0^2 = 0.09 + 0.04
#define VOL0F  0.2f

typedef float v2f __attribute__((ext_vector_type(2)));
typedef float v8f __attribute__((ext_vector_type(8)));

__device__ __forceinline__ float wave_sum32(float v) {
#pragma unroll
  for (int off = 1; off < 32; off <<= 1) v += __shfl_xor(v, off, 32);
  return v;
}

// One workgroup per batch row. All state LDS-resident across the 128 steps.
__global__ void __launch_bounds__(NTH) fp_main_kernel(
    const float* __restrict__ mt,    // mesh_time   (129)
    const float* __restrict__ mesh,  // mesh_space  (2050)
    const float* __restrict__ ic,    // initial_condition (2048)
    const float* __restrict__ bw,    // brownian_increment (128 x 128)
    float* __restrict__ out,         // density (128 x 129 x 2048) ++ cost(2)
    float* __restrict__ ws)          // per-batch cost partials (2 x 128)
{
  extern __shared__ float smem[];
  float* s_xi   = smem + 0 * NXI;
  float* s_w    = smem + 1 * NXI;
  float* s_fc   = smem + 2 * NXI;   // 1/(2*dx_f)
  float* s_bc   = smem + 3 * NXI;   // 1/(2*dx_b)
  float* s_dens = smem + 4 * NXI;
  float* s_dif  = smem + 5 * NXI;
  float* s_A    = smem + 6 * NXI;   // sub
  float* s_B    = smem + 7 * NXI;   // main
  float* s_C    = smem + 8 * NXI;   // sup
  float* s_D    = smem + 9 * NXI;   // rhs
  float* s_red  = smem + 10 * NXI;  // 64 floats of reduction scratch
  float* s_mesh = s_A;              // mesh staging aliases A/B region (2050 <= 2*NXI)

  const int tid  = threadIdx.x;
  const int lane = tid & 31;
  const int wid  = tid >> 5;
  const int b    = blockIdx.x;

  __builtin_prefetch(&bw[b * NSTEP], 0, 3);   // -> global_prefetch_b8
  __builtin_prefetch(&mt[0], 0, 3);

  // ---- stage mesh_space + initial density into LDS (CDNA5 async-to-LDS) ----
#if defined(__HIP_DEVICE_COMPILE__) && __has_builtin(__builtin_amdgcn_global_load_async_to_lds_b32)
  typedef __attribute__((address_space(1))) int gas_t;  // b32 element, global
  typedef __attribute__((address_space(3))) int las_t;  // b32 element, LDS
  for (int i = tid; i < NXI + 2; i += NTH)
    __builtin_amdgcn_global_load_async_to_lds_b32(
        (gas_t*)(uintptr_t)(mesh + i),
        (las_t*)(unsigned)(uintptr_t)(s_mesh + i), 0, 0);
  for (int i = tid; i < NXI; i += NTH)
    __builtin_amdgcn_global_load_async_to_lds_b32(
        (gas_t*)(uintptr_t)(ic + i),
        (las_t*)(unsigned)(uintptr_t)(s_dens + i), 0, 0);
#if __has_builtin(__builtin_amdgcn_s_wait_asynccnt)
  __builtin_amdgcn_s_wait_asynccnt(0);
#else
  asm volatile("s_wait_asynccnt 0x0" ::: "memory");
#endif
#else
  for (int i = tid; i < NXI + 2; i += NTH) s_mesh[i] = mesh[i];
  for (int i = tid; i < NXI; i += NTH) s_dens[i] = ic[i];
#endif
  __syncthreads();

  // per-point mesh constants + write the t=0 density row
  const size_t obase = (size_t)b * (size_t)((NSTEP + 1) * NXI);
#pragma unroll
  for (int j = 0; j < EPT; ++j) {
    const int i = tid + NTH * j;
    const float x0 = s_mesh[i], x1 = s_mesh[i + 1], x2 = s_mesh[i + 2];
    s_xi[i] = x1;
    s_w[i]  = 0.5f * (x2 - x0);
    s_fc[i] = 1.0f / (2.0f * (x2 - x1));
    s_bc[i] = 1.0f / (2.0f * (x1 - x0));
    out[obase + i] = s_dens[i];
  }
  __syncthreads();

  // WMMA identity A-matrix (16x4 f32 layout: V0 = K0 | K2, V1 = K1 | K3)
  // A[m][k] = delta(m,k): selects B row k into D row k -> D acts as a 3x16
  // f32 accumulator bank for the three integrals.
  const v2f Awm = { (lane == 0 || lane == 18) ? 1.0f : 0.0f,
                    (lane == 1 || lane == 19) ? 1.0f : 0.0f };

  float costsum = 0.0f;
  const int rbase = wid * 256;  // each wave owns 256 contiguous points

#pragma unroll 1
  for (int t = 0; t < NSTEP; ++t) {
    const float dtt = mt[t + 1] - mt[t];
    const float dW  = bw[b * NSTEP + t];

    // ---- integrals: fb = sum(max(-x,0) w rho), mean = sum(x w rho),
    //      q2 = sum(x^2 w rho)  via chained V_WMMA_F32_16X16X4_F32 ----
    v8f acc = {};
#pragma unroll 1
    for (int it = 0; it < 16; ++it) {
      const int e   = rbase + it * 16 + (lane & 15);
      const float p = s_dens[e] * s_w[e];
      const float xe = s_xi[e];
      const bool lo = lane < 16;
      // B 4x16 f32 layout: V0 = row0(lanes0-15) | row2(lanes16-31),
      //                    V1 = row1(lanes0-15) | row3(lanes16-31)
      const v2f Bwm = { lo ? fmaxf(-xe, 0.0f) * p : xe * xe * p,
                        lo ? xe * p : 0.0f };
      acc = __builtin_amdgcn_wmma_f32_16x16x4_f32(false, Awm, false, Bwm,
                                                  (short)0, acc, false, false);
    }
    // D row r lives in acc[r] lanes 0-15 (lanes 16-31 = row 8+r = 0)
    const float pf = wave_sum32(acc[0]);
    const float pm = wave_sum32(acc[1]);
    const float pq = wave_sum32(acc[2]);
    if (lane == 0) {
      s_red[wid * 3 + 0] = pf;
      s_red[wid * 3 + 1] = pm;
      s_red[wid * 3 + 2] = pq;
    }
    __syncthreads();
    if (tid == 0) {
      float f = 0.0f, m = 0.0f, q = 0.0f;
      for (int k = 0; k < 8; ++k) {
        f += s_red[k * 3 + 0]; m += s_red[k * 3 + 1]; q += s_red[k * 3 + 2];
      }
      s_red[48] = f; s_red[49] = m; s_red[50] = q;
    }
    __syncthreads();
    const float fb   = s_red[48];
    const float mean = s_red[49];
    const float q2   = s_red[50];
    costsum += dtt * 0.125f * q2;               // integrate(control^2/2, rho)
    const float cs    = fb - 0.1f * mean;       // coeff = cs - 0.4*x
    const float delta = 0.02f * (dW * dW - dtt);
    const float vdW   = VOL0F * dW;

    // ---- first-derivative stencil + tridiagonal coefficients ----
#pragma unroll
    for (int j = 0; j < EPT; ++j) {
      const int i = tid + NTH * j;
      const float dc  = s_dens[i];
      const float dm1 = (i > 0) ? s_dens[i - 1] : 0.0f;
      const float dp1 = (i < NXI - 1) ? s_dens[i + 1] : 0.0f;
      const float fci = s_fc[i], bci = s_bc[i];
      s_dif[i] = fci * (dp1 - dc) + bci * (dc - dm1);

      const float xi    = s_xi[i];
      const float coeff = cs - 0.4f * xi;
      const float inte  = fmaxf(-xi, 0.0f);
      const float invw  = DCOEF / s_w[i];       // D/(ctl*dx) = D*fc/w (bc/w)
      const float coefp = (i < NXI - 1) ? (cs - 0.4f * s_xi[i + 1]) : 0.0f;
      const float coefm = (i > 0)       ? (cs - 0.4f * s_xi[i - 1]) : 0.0f;
      s_C[i] = (i < NXI - 1) ? (-dtt * fci * (coefp + invw)) : 0.0f;
      s_A[i] = (i > 0)       ? ( dtt * bci * (coefm - invw)) : 0.0f;
      s_B[i] = 1.0f + dtt * inte - dtt * (bci - fci) * coeff
             + dtt * invw * (fci + bci);
    }
    __syncthreads();

    // ---- rhs = dens - (VOL0*dW*diff - delta*L[diff]) ----
#pragma unroll
    for (int j = 0; j < EPT; ++j) {
      const int i = tid + NTH * j;
      const float qc = s_dif[i];
      const float qm = (i > 0) ? s_dif[i - 1] : 0.0f;
      const float qp = (i < NXI - 1) ? s_dif[i + 1] : 0.0f;
      const float d2 = delta * (s_fc[i] * (qp - qc) + s_bc[i] * (qc - qm));
      s_D[i] = s_dens[i] - (vdW * qc - d2);
    }
    __syncthreads();

    // ---- parallel cyclic reduction (11 levels, in place) ----
#pragma unroll 1
    for (int s = 1; s < NXI; s <<= 1) {
      float na[EPT], nb[EPT], nc[EPT], nd[EPT];
#pragma unroll
      for (int j = 0; j < EPT; ++j) {
        const int i = tid + NTH * j;
        const float ai = s_A[i], bi = s_B[i], ci = s_C[i], di = s_D[i];
        const int im = i - s, ip = i + s;
        float am = 0.0f, bm = 1.0f, cm = 0.0f, dm = 0.0f;
        float ap = 0.0f, bp = 1.0f, cp = 0.0f, dp = 0.0f;
        if (im >= 0)  { am = s_A[im]; bm = s_B[im]; cm = s_C[im]; dm = s_D[im]; }
        if (ip < NXI) { ap = s_A[ip]; bp = s_B[ip]; cp = s_C[ip]; dp = s_D[ip]; }
        const float k1 = ai / bm;
        const float k2 = ci / bp;
        na[j] = -am * k1;
        nc[j] = -cp * k2;
        nb[j] = bi - cm * k1 - ap * k2;
        nd[j] = di - dm * k1 - dp * k2;
      }
      __syncthreads();
#pragma unroll
      for (int j = 0; j < EPT; ++j) {
        const int i = tid + NTH * j;
        s_A[i] = na[j]; s_B[i] = nb[j]; s_C[i] = nc[j]; s_D[i] = nd[j];
      }
      __syncthreads();
    }

    // fully decoupled: x_i = d_i / b_i ; commit new density + trajectory row
    float* orow = out + obase + (size_t)(t + 1) * NXI;
#pragma unroll
    for (int j = 0; j < EPT; ++j) {
      const int i = tid + NTH * j;
      const float x = s_D[i] / s_B[i];
      s_dens[i] = x;
      orow[i] = x;
    }
    __syncthreads();
  }

  // terminal integral: sum(x^2 * w * rho_final)
  float accT = 0.0f;
#pragma unroll
  for (int j = 0; j < EPT; ++j) {
    const int i = tid + NTH * j;
    accT += s_xi[i] * s_xi[i] * s_w[i] * s_dens[i];
  }
  accT = wave_sum32(accT);
  if (lane == 0) s_red[wid] = accT;
  __syncthreads();
  if (tid == 0) {
    float tsum = 0.0f;
    for (int k = 0; k < 8; ++k) tsum += s_red[k];
    ws[2 * b + 0] = costsum;
    ws[2 * b + 1] = tsum;
  }
}

// Fold the 128 per-batch partials into the two cost outputs (deterministic).
__global__ void fp_cost_kernel(const float* __restrict__ ws,
                               float* __restrict__ out) {
  if (blockIdx.x == 0 && threadIdx.x == 0) {
    float c0 = 0.0f, c1 = 0.0f;
    for (int k = 0; k < NBATCH; ++k) { c0 += ws[2 * k]; c1 += ws[2 * k + 1]; }
    const size_t off = (size_t)NBATCH * (size_t)((NSTEP + 1) * NXI);
    out[off + 0] = c0 * (1.0f / NBATCH);
    out[off + 1] = c1 * (1.0f / NBATCH);
  }
}

extern "C" void kernel_launch(void* const* d_in, const int* in_sizes, int n_in,
                              void* d_out, int out_size, void* d_ws, size_t ws_size,
                              hipStream_t stream) {
  (void)in_sizes; (void)n_in; (void)out_size; (void)ws_size;
  const float* mt   = (const float*)d_in[0];
  const float* mesh = (const float*)d_in[1];
  const float* ic   = (const float*)d_in[2];
  const float* bw   = (const float*)d_in[3];
  float* out = (float*)d_out;
  float* ws  = (float*)d_ws;

  const size_t smem_bytes = (size_t)(10 * NXI + 64) * sizeof(float); // ~82 KB
  fp_main_kernel<<<NBATCH, NTH, smem_bytes, stream>>>(mt, mesh, ic, bw, out, ws);
  fp_cost_kernel<<<1, 32, 0, stream>>>(ws, out);
}